// CoupledClustersLoss_71502615544388
// MI455X (gfx1250) — compile-verified
//
#include <hip/hip_runtime.h>
#include <hip/hip_bf16.h>

#define N_CLASSES 1024
#define N_SAMPLES 32
#define DIM 512
#define MARGIN 0.3f

#define CHUNK 128
#define LDSTRIDE (CHUNK + 4)   // 132 floats; 132 % 64 == 4 -> conflict-free banked access

typedef float v2f __attribute__((ext_vector_type(2)));
typedef float v8f __attribute__((ext_vector_type(8)));

// One workgroup (256 threads = 8 waves) per class.
// Computes G = X * X^T restricted to pos columns with V_WMMA_F32_16X16X4_F32,
// then the coupled-clusters loss for the class via the Gram expansion.
__global__ __launch_bounds__(256) void ccl_per_class(const float* __restrict__ emb,
                                                     float* __restrict__ losses) {
    __shared__ __align__(16) float Xs[64 * LDSTRIDE];
    __shared__ float rs[64][2];    // partial pos-row-sums of G, per column group
    __shared__ float diag[64];     // ||x_i||^2

    const int c    = blockIdx.x;
    const int t    = threadIdx.x;
    const int lane = t & 31;
    const int wave = t >> 5;

    const float* base = emb + (size_t)c * 64 * DIM;

    // Tile assignment: wave w -> A row-group ri in 0..3, B (pos) col-group rj in 0..1
    const int ri   = wave >> 1;
    const int rj   = wave & 1;
    const int m    = lane & 15;
    const int half = lane >> 4;            // 0: K 0..1, 1: K 2..3
    const int rowA = ri * 16 + m;
    const int rowB = rj * 16 + m;          // B fragment of X^T == A fragment of rows rj*16..+15

    v8f acc = {};                          // 16x16 f32 accumulator tile
    float dsum = 0.0f;                     // partial ||x||^2: 4 threads per row
    const int drow  = t >> 2;
    const int dcol0 = (t & 3) * 32;

    for (int ch = 0; ch < 4; ++ch) {
        __syncthreads();
        // Stage 64 x 128 f32 chunk, coalesced float4 loads, padded LDS rows
        const float* gsrc = base + ch * CHUNK;
        #pragma unroll
        for (int it = 0; it < 8; ++it) {
            int idx = it * 256 + t;                 // 2048 float4 slots
            int r   = idx >> 5;                     // 32 float4 per row
            int c4  = idx & 31;
            float4 v = *(const float4*)(gsrc + (size_t)r * DIM + c4 * 4);
            *(float4*)(Xs + r * LDSTRIDE + c4 * 4) = v;
        }
        __syncthreads();

        // Gram tile accumulation: 32 WMMA f32 16x16x4 per chunk per wave
        const float* pa = Xs + rowA * LDSTRIDE + 2 * half;
        const float* pb = Xs + rowB * LDSTRIDE + 2 * half;
        #pragma unroll
        for (int k = 0; k < CHUNK; k += 4) {
            v2f a = *(const v2f*)(pa + k);
            v2f b = *(const v2f*)(pb + k);
            acc = __builtin_amdgcn_wmma_f32_16x16x4_f32(
                false, a, false, b, (short)0, acc, false, false);
        }

        // Diagonal (norm^2) partials on VALU
        const float* pd = Xs + drow * LDSTRIDE + dcol0;
        #pragma unroll
        for (int j = 0; j < 32; ++j) {
            float x = pd[j];
            dsum = fmaf(x, x, dsum);
        }
    }

    // Row-sum each accumulator VGPR across the 16 columns it holds.
    // C layout: VGPR r, lanes 0-15 -> row r cols 0-15 ; lanes 16-31 -> row r+8 cols 0-15.
    #pragma unroll
    for (int r = 0; r < 8; ++r) {
        float v = acc[r];
        v += __shfl_xor(v, 1);
        v += __shfl_xor(v, 2);
        v += __shfl_xor(v, 4);
        v += __shfl_xor(v, 8);
        if (m == 0) {  // lanes 0 and 16 hold the two half-tile row sums
            rs[ri * 16 + half * 8 + r][rj] = v;
        }
    }

    // Reduce 4 diag partials per row (fixed shuffle order -> deterministic)
    {
        float v = dsum;
        v += __shfl_xor(v, 1);
        v += __shfl_xor(v, 2);
        if ((t & 3) == 0) diag[drow] = v;
    }
    __syncthreads();

    // Final per-class math on wave 0: lane i covers pos_i and neg_i
    if (wave == 0) {
        int i = lane;
        float sp = (rs[i][0]      + rs[i][1])      * (1.0f / 32.0f);  // a . pos_i
        float sn = (rs[32 + i][0] + rs[32 + i][1]) * (1.0f / 32.0f);  // a . neg_i
        float sumS = sp;
        sumS += __shfl_xor(sumS, 1);
        sumS += __shfl_xor(sumS, 2);
        sumS += __shfl_xor(sumS, 4);
        sumS += __shfl_xor(sumS, 8);
        sumS += __shfl_xor(sumS, 16);
        float aa = sumS * (1.0f / 32.0f);                              // ||a||^2
        float ap = aa - 2.0f * sp + diag[i];
        float nd = aa - 2.0f * sn + diag[32 + i];
        float an = nd;
        an = fminf(an, __shfl_xor(an, 1));
        an = fminf(an, __shfl_xor(an, 2));
        an = fminf(an, __shfl_xor(an, 4));
        an = fminf(an, __shfl_xor(an, 8));
        an = fminf(an, __shfl_xor(an, 16));
        float li = fmaxf(ap - an + MARGIN, 0.0f);
        li += __shfl_xor(li, 1);
        li += __shfl_xor(li, 2);
        li += __shfl_xor(li, 4);
        li += __shfl_xor(li, 8);
        li += __shfl_xor(li, 16);
        if (lane == 0) losses[c] = li;
    }
}

__global__ __launch_bounds__(256) void ccl_mean(const float* __restrict__ losses,
                                                float* __restrict__ out) {
    __shared__ float red[256];
    int t = threadIdx.x;
    float v = losses[t] + losses[t + 256] + losses[t + 512] + losses[t + 768];
    red[t] = v;
    __syncthreads();
    for (int s = 128; s > 0; s >>= 1) {
        if (t < s) red[t] += red[t + s];
        __syncthreads();
    }
    if (t == 0) out[0] = red[0] * (1.0f / (float)N_CLASSES);
}

extern "C" void kernel_launch(void* const* d_in, const int* in_sizes, int n_in,
                              void* d_out, int out_size, void* d_ws, size_t ws_size,
                              hipStream_t stream) {
    (void)in_sizes; (void)n_in; (void)out_size; (void)ws_size;
    const float* emb = (const float*)d_in[0];   // (2*1024*32, 512) f32
    // d_in[1] (target) is unused by the reference computation.
    float* losses = (float*)d_ws;               // 1024 f32 scratch
    float* out    = (float*)d_out;              // scalar f32

    ccl_per_class<<<N_CLASSES, 256, 0, stream>>>(emb, losses);
    ccl_mean<<<1, 256, 0, stream>>>(losses, out);
}